// HMCSampler_17300128269077
// MI455X (gfx1250) — compile-verified
//
#include <hip/hip_runtime.h>
#include <stdint.h>

// ---- static problem shape / HMC hyperparameters -------------------------
#define D       16
#define BATCH   1024
#define NSAMP   32
#define NTOT    (BATCH * NSAMP)          // 32768 chains
#define LSTEP   3
#define NRUN    3
#define GAMMA   0.05f
#define EPS_JIT 1e-3f
#define BOUNDS2 1.0e6f                   // 1000^2
#define WPB     8                        // waves (chains) per workgroup
#define BLOCK   (WPB * 32)

#define LOG2E   1.4426950408889634f
#define LN2     0.6931471805599453f

typedef __attribute__((ext_vector_type(2))) float v2f;
typedef __attribute__((ext_vector_type(8))) float v8f;

// fast hardware transcendentals (v_rcp/v_rsq/v_sqrt/v_exp/v_log _f32)
__device__ __forceinline__ float frcp(float x)  { return __builtin_amdgcn_rcpf(x); }
__device__ __forceinline__ float frsq(float x)  { return __builtin_amdgcn_rsqf(x); }
__device__ __forceinline__ float fsqrt(float x) { return __builtin_amdgcn_sqrtf(x); }
__device__ __forceinline__ float fexp2(float x) { return __builtin_amdgcn_exp2f(x); }
__device__ __forceinline__ float flog2(float x) { return __builtin_amdgcn_logf(x); }

// ---- counter-based RNG ---------------------------------------------------
__device__ __forceinline__ uint64_t splitmix64(uint64_t x) {
  x += 0x9E3779B97F4A7C15ULL;
  x = (x ^ (x >> 30)) * 0xBF58476D1CE4E5B9ULL;
  x = (x ^ (x >> 27)) * 0x94D049BB133111EBULL;
  return x ^ (x >> 31);
}
__device__ __forceinline__ float u01(uint32_t r) {        // (0,1]
  return ((float)r + 1.0f) * 2.3283064365386963e-10f;
}
// sum across a 16-lane half (halves hold duplicated values -> true sum)
__device__ __forceinline__ float hsum16(float x) {
  x += __shfl_xor(x, 1);
  x += __shfl_xor(x, 2);
  x += __shfl_xor(x, 4);
  x += __shfl_xor(x, 8);
  return x;
}

struct Eval { float g; float U; };

// One metric evaluation for the wave's chain: s=sigmoid(zW+c),
// M = B diag(s) B^T + eps I  (4x V_WMMA_F32_16X16X4_F32),
// Cholesky -> U = -sum log Lii, grad via diag(B^T M^-1 B).
__device__ __forceinline__ Eval metric_eval(
    int w, int j, int half,
    const float (&Bsh)[D][D], const float (&Wsh)[D][D], const float (&csh)[D],
    float (&zsh)[WPB][D], float (&ssh)[WPB][D], float (&ush)[WPB][D],
    float (&dish)[WPB][D], float (&Msh)[WPB][D][D + 1])
{
  // ---- s_j = sigmoid(c_j + sum_k z_k W[k][j]); lane -> feature j ----
  float t = csh[j];
  #pragma unroll
  for (int k = 0; k < D; ++k) t = fmaf(zsh[w][k], Wsh[k][j], t);
  const float s = frcp(1.0f + fexp2(-t * LOG2E));
  ssh[w][j] = s;
  __builtin_amdgcn_wave_barrier();

  // ---- M = (B*diag(s)) @ B^T via 4 chained f32 WMMAs (K = 16) ----
  // A 16x4 frag: lane row = j; VGPRs = K pair (halves carry K+0/1 vs K+2/3).
  // B^T 4x16 frag: lane col = j; same K striping.
  v8f acc = {0.f, 0.f, 0.f, 0.f, 0.f, 0.f, 0.f, 0.f};
  #pragma unroll
  for (int cb = 0; cb < 4; ++cb) {
    const int k0 = cb * 4 + half * 2;
    const float b0 = Bsh[j][k0], b1 = Bsh[j][k0 + 1];
    v2f a, bt;
    a.x  = b0 * ssh[w][k0];
    a.y  = b1 * ssh[w][k0 + 1];
    bt.x = b0;                      // B^T[k][j] == B[j][k]
    bt.y = b1;
    acc = __builtin_amdgcn_wmma_f32_16x16x4_f32(
        false, a, false, bt, (short)0, acc, false, false);
  }
  // C layout: lane col=j, VGPR v -> row v + 8*half. Scatter to LDS, +eps on diag.
  #pragma unroll
  for (int v = 0; v < 8; ++v) {
    const int row = v + half * 8;
    Msh[w][row][j] = acc[v] + ((row == j) ? EPS_JIT : 0.0f);
  }
  __builtin_amdgcn_wave_barrier();

  // ---- in-place Cholesky, lane owns row j (halves duplicate identically).
  // rs = 1/sqrt(d) once per column; Ljj = d*rs; 1/Ljj = rs (no divides).
  // Diagonal is never written back: lane captures its Ljj and rs in registers.
  float myLjj = 1.0f, myrs = 1.0f;
  for (int jc = 0; jc < D; ++jc) {
    const float dcur = Msh[w][jc][jc];
    const float rs   = frsq(dcur);
    if (j == jc) { myLjj = dcur * rs; myrs = rs; }
    if (j > jc) {
      const float Lij = Msh[w][j][jc] * rs;
      Msh[w][j][jc] = Lij;
      for (int kc = jc + 1; kc <= j; ++kc)
        Msh[w][j][kc] -= Lij * Msh[w][kc][jc];
    }
    __builtin_amdgcn_wave_barrier();
  }
  dish[w][j] = myrs;                         // 1/L_jj for the triangular solve
  __builtin_amdgcn_wave_barrier();

  // U = -0.5 * logdet(M) = -sum_i log L_ii
  const float U = -LN2 * hsum16(flog2(myLjj));

  // ---- forward solve L Y = B (lane owns RHS column j); q_j = ||Y[:,j]||^2 ----
  float y[D];
  float q = 0.0f;
  #pragma unroll
  for (int i = 0; i < D; ++i) {
    float r = Bsh[i][j];
    #pragma unroll
    for (int kk = 0; kk < i; ++kk) r = fmaf(-Msh[w][i][kk], y[kk], r);
    y[i] = r * dish[w][i];
    q = fmaf(y[i], y[i], q);
  }
  ush[w][j] = s * (1.0f - s) * q;
  __builtin_amdgcn_wave_barrier();

  // grad_j = -0.5 * sum_k u_k W[j][k]
  float g = 0.0f;
  #pragma unroll
  for (int k = 0; k < D; ++k) g = fmaf(ush[w][k], Wsh[j][k], g);
  Eval e; e.g = -0.5f * g; e.U = U;
  return e;
}

__global__ __launch_bounds__(BLOCK) void hmc_fused(
    const float* __restrict__ z0, const float* __restrict__ Wg,
    const float* __restrict__ cg, const float* __restrict__ Bg,
    float* __restrict__ out)
{
  __shared__ float Bsh[D][D], Wsh[D][D], csh[D];
  __shared__ float zsh[WPB][D], ssh[WPB][D], ush[WPB][D], dish[WPB][D];
  __shared__ float Msh[WPB][D][D + 1];

  const int tid = threadIdx.x;
  if (tid < D * D) {
    Bsh[tid >> 4][tid & 15] = Bg[tid];
    Wsh[tid >> 4][tid & 15] = Wg[tid];
  }
  if (tid < D) csh[tid] = cg[tid];
  __syncthreads();

  const int w    = tid >> 5;
  const int lane = tid & 31;
  const int j    = lane & 15;       // element index; halves mirror each other
  const int half = lane >> 4;
  const int sidx = blockIdx.x * WPB + w;        // chain id in [0, 32768)
  const int brow = sidx & (BATCH - 1);          // z is z_0 tiled (NSAMP,1)

  float zj = z0[brow * D + j];
  zsh[w][j] = zj;
  __builtin_amdgcn_wave_barrier();

  for (int it = 0; it < NRUN; ++it) {
    Eval e = metric_eval(w, j, half, Bsh, Wsh, csh, zsh, ssh, ush, dish, Msh);

    // momentum v ~ N(0, STD0^2=1): Box-Muller from counter-based RNG
    const uint64_t r =
        splitmix64(((uint64_t)(it + 1) << 40) ^ ((uint64_t)sidx << 8) ^ (uint64_t)j);
    const float u1 = u01((uint32_t)(r >> 32));
    const float u2 = u01((uint32_t)r);
    float vj = fsqrt(-2.0f * LN2 * flog2(u1)) * __cosf(6.28318530718f * u2);

    const float H0   = e.U + 0.5f * hsum16(vj * vj);
    const float zold = zj;
    float gj = e.g;

    // tempered leapfrog; scales = prev_beta/beta = {4.0, 0.9, 10/13}
    #pragma unroll
    for (int k = 0; k < LSTEP; ++k) {
      vj = fmaf(-0.5f * GAMMA, gj, vj);
      zj = fmaf(GAMMA, vj, zj);
      zsh[w][j] = zj;
      __builtin_amdgcn_wave_barrier();
      e  = metric_eval(w, j, half, Bsh, Wsh, csh, zsh, ssh, ush, dish, Msh);
      gj = e.g;
      vj = fmaf(-0.5f * GAMMA, gj, vj);
      vj *= (k == 0) ? 4.0f : ((k == 1) ? 0.9f : 0.769230769231f);
    }

    // Metropolis accept (wave-uniform decision)
    const float H1  = e.U + 0.5f * hsum16(vj * vj);
    const float zn2 = hsum16(zj * zj);
    float alpha = fminf(1.0f, fexp2((H0 - H1) * LOG2E));
    if (zn2 > BOUNDS2) alpha = 0.0f;
    const uint32_t ra = (uint32_t)(splitmix64(
        0xD1B54A32D192ED03ULL ^ ((uint64_t)(it + 1) << 32) ^ (uint64_t)sidx) >> 32);
    const bool accept = (u01(ra) < alpha);
    if (!accept) zj = zold;
    zsh[w][j] = zj;
    __builtin_amdgcn_wave_barrier();
  }

  // output layout: (b, n, d) with chain sidx = nidx*BATCH + bidx
  if (half == 0) {
    const int nidx = sidx >> 10;
    const int bidx = sidx & (BATCH - 1);
    out[(bidx * NSAMP + nidx) * D + j] = zj;
  }
}

extern "C" void kernel_launch(void* const* d_in, const int* in_sizes, int n_in,
                              void* d_out, int out_size, void* d_ws, size_t ws_size,
                              hipStream_t stream) {
  (void)in_sizes; (void)n_in; (void)d_ws; (void)ws_size; (void)out_size;
  const float* z0 = (const float*)d_in[0];
  const float* W  = (const float*)d_in[1];
  const float* c  = (const float*)d_in[2];
  const float* Bm = (const float*)d_in[3];
  hmc_fused<<<NTOT / WPB, BLOCK, 0, stream>>>(z0, W, c, Bm, (float*)d_out);
}